// MinibatchDiscrimination_34505767256334
// MI455X (gfx1250) — compile-verified
//
#include <hip/hip_runtime.h>
#include <hip/hip_bf16.h>

typedef __attribute__((ext_vector_type(2))) float v2f;
typedef __attribute__((ext_vector_type(8))) float v8f;

#define B_SZ   256
#define IN_F   512
#define K_INT  75
#define OUT_F  64
#define NCOL   (K_INT * OUT_F)   /* 4800 */
#define JC     8                 /* j-chunks */
#define JCHUNK (B_SZ / JC)       /* 32 */

// ---------------------------------------------------------------------------
// Phase 1: M = x @ reshape(T, [512, 4800]) via V_WMMA_F32_16X16X4_F32.
// Output stored TRANSPOSED: Mt[o][k][b] (b contiguous) for coalesced phase 2.
// Grid: 600 blocks x 256 threads = 4800 waves = 4800 16x16 output tiles.
// ---------------------------------------------------------------------------
__global__ __launch_bounds__(256) void mbdisc_gemm_wmma(
    const float* __restrict__ x,      // [256, 512]
    const float* __restrict__ T,      // [512, 4800]
    float* __restrict__ Mt)           // [64][75][256]
{
    const int tid  = threadIdx.x;
    const int lane = tid & 31;
    const int wave = tid >> 5;
    const int tile = blockIdx.x * 8 + wave;   // 0..4799
    const int tM   = tile & 15;               // 16 row tiles
    const int tN   = tile >> 4;               // 300 col tiles

    const int m     = lane & 15;              // A row / B,C col within tile
    const int khalf = (lane >> 4) * 2;        // K sub-offset {0,2}
    const int n     = tN * 16 + m;            // global column

    // A fragment pointer: x[tM*16 + m][kk*4 + khalf + {0,1}]  (8B aligned)
    const float* arow = x + (size_t)(tM * 16 + m) * IN_F + khalf;
    // B fragment pointer: T[kk*4 + khalf + {0,1}][n]
    const float* bcol = T + (size_t)khalf * NCOL + n;

    v8f c = {};
    #pragma unroll 4
    for (int kk = 0; kk < IN_F / 4; ++kk) {
        v2f a = *(const v2f*)arow;            // A[m][k], A[m][k+1]
        v2f b;
        b.x = bcol[0];                        // B[k][n]
        b.y = bcol[NCOL];                     // B[k+1][n]
        arow += 4;
        bcol += (size_t)4 * NCOL;
        // (neg_a, A, neg_b, B, c_mod, C, reuse_a, reuse_b)
        c = __builtin_amdgcn_wmma_f32_16x16x4_f32(
                false, a, false, b, (short)0, c, false, false);
    }

    // D layout: VGPR v, lanes 0-15 -> row v, lanes 16-31 -> row v+8; col = lane&15
    const int b0   = tM * 16 + (lane >> 4) * 8;   // batch row base
    const int o    = n & 63;
    const int kint = n >> 6;
    float* dst = Mt + ((size_t)o * K_INT + kint) * B_SZ;
    #pragma unroll
    for (int v = 0; v < 8; ++v)
        dst[b0 + v] = c[v];
}

// ---------------------------------------------------------------------------
// Phase 2: partial[jc][i][o] = sum_{j in chunk jc} exp(-sum_k |M_i - M_j|)
// One block per (o, jc). Thread i keeps its 75-elem row in VGPRs; the 32-row
// j-chunk lives in LDS (9.6 KB); inner reads are LDS broadcasts.
// ---------------------------------------------------------------------------
__global__ __launch_bounds__(256) void mbdisc_pairwise(
    const float* __restrict__ Mt,     // [64][75][256]
    float* __restrict__ part)         // [JC][256][64]
{
    __shared__ float lds[K_INT * JCHUNK];     // 2400 floats

    const int i  = threadIdx.x;
    const int o  = blockIdx.x;
    const int jc = blockIdx.y;
    const int j0 = jc * JCHUNK;

    const float* base = Mt + (size_t)o * K_INT * B_SZ;

    // Own row -> registers (coalesced: consecutive i contiguous)
    float mi[K_INT];
    #pragma unroll
    for (int k = 0; k < K_INT; ++k)
        mi[k] = base[(size_t)k * B_SZ + i];

    // j-chunk -> LDS (32-contiguous segments)
    for (int p = i; p < K_INT * JCHUNK; p += 256) {
        int k  = p / JCHUNK;
        int jj = p - k * JCHUNK;
        lds[p] = base[(size_t)k * B_SZ + j0 + jj];
    }
    __syncthreads();

    float acc = 0.0f;
    for (int jj = 0; jj < JCHUNK; ++jj) {
        float dist = 0.0f;
        #pragma unroll
        for (int k = 0; k < K_INT; ++k)
            dist += __builtin_fabsf(mi[k] - lds[k * JCHUNK + jj]);
        acc += __expf(-dist);
    }
    part[((size_t)jc * B_SZ + i) * OUT_F + o] = acc;
}

// ---------------------------------------------------------------------------
// Phase 3: deterministic reduction over j-chunks; -1 removes self-distance.
// ---------------------------------------------------------------------------
__global__ __launch_bounds__(256) void mbdisc_reduce(
    const float* __restrict__ part,   // [JC][256][64]
    float* __restrict__ out)          // [256][64]
{
    const int idx = blockIdx.x * 256 + threadIdx.x;   // i*64+o, 16384 total
    float s = -1.0f;
    #pragma unroll
    for (int jc = 0; jc < JC; ++jc)
        s += part[(size_t)jc * (B_SZ * OUT_F) + idx];
    out[idx] = s;
}

extern "C" void kernel_launch(void* const* d_in, const int* in_sizes, int n_in,
                              void* d_out, int out_size, void* d_ws, size_t ws_size,
                              hipStream_t stream) {
    const float* x = (const float*)d_in[0];   // [256, 512]
    const float* T = (const float*)d_in[1];   // [512, 75, 64] == [512, 4800]
    float* out = (float*)d_out;               // [256, 64]

    float* Mt   = (float*)d_ws;                                   // 4,915,200 B
    float* part = (float*)((char*)d_ws + (size_t)OUT_F * K_INT * B_SZ * sizeof(float));

    // Phase 1: 4800 tiles / 8 waves per block = 600 blocks
    mbdisc_gemm_wmma<<<600, 256, 0, stream>>>(x, T, Mt);

    // Phase 2: (o, jc) grid = 64 x 8 = 512 blocks
    mbdisc_pairwise<<<dim3(OUT_F, JC), 256, 0, stream>>>(Mt, part);

    // Phase 3: 16384 outputs
    mbdisc_reduce<<<(B_SZ * OUT_F) / 256, 256, 0, stream>>>(part, out);
}